// MessagePassingLayer_44126493999752
// MI455X (gfx1250) — compile-verified
//
#include <hip/hip_runtime.h>

#define BB 16
#define PP 2048
#define CC 2048
#define HH 32
#define CHUNK 128            // rows staged per step
#define NPAIR (CHUNK / 2)    // 64 row-pairs
#define LSTRIDE 66           // floats per row-pair in LDS (64 + 2 bank skew)
#define NW 4                 // waves per workgroup (128 threads)

typedef __attribute__((ext_vector_type(2))) float v2f;
typedef __attribute__((ext_vector_type(4))) float v4f;
typedef __attribute__((ext_vector_type(8))) float v8f;

__device__ __forceinline__ v8f wmma_f32(v2f a, v2f b, v8f c) {
  // D = A(16x4) * B(4x16) + C, fp32 exact
  return __builtin_amdgcn_wmma_f32_16x16x4_f32(false, a, false, b,
                                               (short)0, c, false, false);
}

__device__ __forceinline__ float fast_sigmoid(float x) {
  return 1.0f / (1.0f + __expf(-x));
}
__device__ __forceinline__ float fast_tanh(float x) {
  float e = __expf(-2.0f * x);
  return (1.0f - e) / (1.0f + e);
}

// ---- staging: global chunk (CHUNK x 32, contiguous) -> row-pair-interleaved LDS
// LDS element (p, h) lives at buf[(p>>1)*LSTRIDE + 2*h + (p&1)], so a B operand
// (k, k+1 in consecutive VGPRs) is a single aligned ds_load_b64.
__device__ __forceinline__ void load_chunk(const float* __restrict__ src_chunk,
                                           int tid, v4f r0[4], v4f r1[4]) {
  const int q = tid >> 1;        // row pair
  const int ht = tid & 1;        // half of the 32 columns
  const float* rowA = src_chunk + (2 * q) * HH + 16 * ht;
#pragma unroll
  for (int j = 0; j < 4; ++j) {
    r0[j] = *(const v4f*)(rowA + 4 * j);
    r1[j] = *(const v4f*)(rowA + HH + 4 * j);
  }
}

__device__ __forceinline__ void store_chunk(float* buf, int tid,
                                            const v4f r0[4], const v4f r1[4]) {
  const int q = tid >> 1;
  const int ht = tid & 1;
  float* dst = buf + q * LSTRIDE + 32 * ht;
#pragma unroll
  for (int j = 0; j < 4; ++j) {
#pragma unroll
    for (int e = 0; e < 4; ++e) {
      v2f p;
      p.x = r0[j][e];
      p.y = r1[j][e];
      *(v2f*)(dst + (4 * j + e) * 2) = p;  // 8B-aligned ds_store_b64
    }
  }
}

// ---- fused GRUCell for one 16-row tile (acc = hidden h, D layout) ----
__device__ __forceinline__ void gru_tile(
    v8f acc0, v8f acc1,
    const float* __restrict__ x_rows,
    const float* msg_lds,
    const float* __restrict__ w_ih, const float* __restrict__ w_hh,
    const float* __restrict__ b_ih, const float* __restrict__ b_hh,
    float* __restrict__ out_rows, int lane) {
  const int lo = lane & 15;
  const int hi = lane >> 4;

  // r,z gate tiles (j=0..63) share accumulators (i_*+h_* is a plain sum);
  // n gate keeps i_n / h_n separate because of r * h_n.
  v8f accs[4], gin[2], ghn[2];
#pragma unroll
  for (int t = 0; t < 4; ++t) {
    float bv = b_ih[16 * t + lo] + b_hh[16 * t + lo];
    v8f v;
#pragma unroll
    for (int r = 0; r < 8; ++r) v[r] = bv;
    accs[t] = v;
  }
#pragma unroll
  for (int t = 0; t < 2; ++t) {
    float bi = b_ih[64 + 16 * t + lo];
    float bh = b_hh[64 + 16 * t + lo];
    v8f vi, vh;
#pragma unroll
    for (int r = 0; r < 8; ++r) { vi[r] = bi; vh[r] = bh; }
    gin[t] = vi;
    ghn[t] = vh;
  }

#pragma unroll
  for (int ks = 0; ks < 8; ++ks) {
    const int kk = 4 * ks + 2 * hi;  // even -> 8B aligned
    v2f ax = *(const v2f*)(x_rows + lo * HH + kk);
    v2f ah = *(const v2f*)(msg_lds + lo * HH + kk);
#pragma unroll
    for (int t = 0; t < 4; ++t) {
      v2f bi = *(const v2f*)(w_ih + (16 * t + lo) * HH + kk);
      v2f bh = *(const v2f*)(w_hh + (16 * t + lo) * HH + kk);
      accs[t] = wmma_f32(ax, bi, accs[t]);
      accs[t] = wmma_f32(ah, bh, accs[t]);
    }
#pragma unroll
    for (int t = 0; t < 2; ++t) {
      v2f bi = *(const v2f*)(w_ih + (64 + 16 * t + lo) * HH + kk);
      v2f bh = *(const v2f*)(w_hh + (64 + 16 * t + lo) * HH + kk);
      gin[t] = wmma_f32(ax, bi, gin[t]);
      ghn[t] = wmma_f32(ah, bh, ghn[t]);
    }
  }

#pragma unroll
  for (int ht = 0; ht < 2; ++ht) {
    v8f hmsg = ht ? acc1 : acc0;
#pragma unroll
    for (int r = 0; r < 8; ++r) {
      float rg = fast_sigmoid(accs[ht][r]);
      float zg = fast_sigmoid(accs[2 + ht][r]);
      float ng = fast_tanh(gin[ht][r] + rg * ghn[ht][r]);
      float hv = hmsg[r];
      float nv = (1.0f - zg) * ng + zg * hv;
      int m = r + 8 * hi;
      out_rows[m * HH + 16 * ht + lo] = nv;
    }
  }
}

// Phase 1: channel_msg[b,c,:] = sum_p adj[b,p,c] * path[b,p,:]  then GRU_c.
__global__ __launch_bounds__(128) void k_channel(
    const float* __restrict__ path_states, const float* __restrict__ channel_states,
    const float* __restrict__ adj,
    const float* __restrict__ w_ih_c, const float* __restrict__ w_hh_c,
    const float* __restrict__ b_ih_c, const float* __restrict__ b_hh_c,
    float* __restrict__ new_channel) {
  __shared__ float lds_buf[2][NPAIR * LSTRIDE];
  __shared__ float lds_msg[NW][16 * HH];

  const int tid = threadIdx.x;
  const int wid = tid >> 5;
  const int lane = tid & 31;
  const int lo = lane & 15, hi = lane >> 4;
  const int b = blockIdx.y;
  const int c0 = blockIdx.x * (NW * 16) + wid * 16;

  const float* adj_col = adj + (size_t)b * PP * CC + c0 + lo;  // A: m=c across lanes
  const float* path_b = path_states + (size_t)b * PP * HH;

  v4f r0[4], r1[4];
  load_chunk(path_b, tid, r0, r1);

  v8f acc0 = {}, acc1 = {};
  const int NCH = PP / CHUNK;  // 16
  for (int ci = 0; ci < NCH; ++ci) {
    float* buf = lds_buf[ci & 1];
    store_chunk(buf, tid, r0, r1);
    __syncthreads();  // double buffer -> single barrier per chunk is sufficient
    if (ci + 1 < NCH)
      load_chunk(path_b + (size_t)(ci + 1) * CHUNK * HH, tid, r0, r1);

    const int ch = ci * CHUNK;
#pragma unroll 8
    for (int k4 = 0; k4 < CHUNK / 4; ++k4) {
      v2f a;  // A[m][k] = adj[p0+k][c0+m]; k = vgpr + 2*hi
      a.x = adj_col[(size_t)(ch + 4 * k4 + 2 * hi) * CC];
      a.y = adj_col[(size_t)(ch + 4 * k4 + 2 * hi + 1) * CC];
      const float* bp = buf + (2 * k4 + hi) * LSTRIDE + 2 * lo;
      v2f b0 = *(const v2f*)(bp);        // k,k+1 contiguous -> ds_load_b64
      v2f b1 = *(const v2f*)(bp + 32);   // h 16..31 tile
      acc0 = wmma_f32(a, b0, acc0);
      acc1 = wmma_f32(a, b1, acc1);
    }
  }

  float* msg = (float*)lds_msg[wid];
#pragma unroll
  for (int r = 0; r < 8; ++r) {
    const int m = r + 8 * hi;
    msg[m * HH + lo] = acc0[r];
    msg[m * HH + 16 + lo] = acc1[r];
  }
  gru_tile(acc0, acc1, channel_states + ((size_t)b * CC + c0) * HH, msg,
           w_ih_c, w_hh_c, b_ih_c, b_hh_c,
           new_channel + ((size_t)b * CC + c0) * HH, lane);
}

// Phase 2: path_msg[b,p,:] = sum_c adj[b,p,c] * new_channel[b,c,:]  then GRU_p.
__global__ __launch_bounds__(128) void k_path(
    const float* __restrict__ path_states, const float* __restrict__ adj,
    const float* __restrict__ new_channel,
    const float* __restrict__ w_ih_p, const float* __restrict__ w_hh_p,
    const float* __restrict__ b_ih_p, const float* __restrict__ b_hh_p,
    float* __restrict__ new_path) {
  __shared__ float lds_buf[2][NPAIR * LSTRIDE];
  __shared__ float lds_msg[NW][16 * HH];

  const int tid = threadIdx.x;
  const int wid = tid >> 5;
  const int lane = tid & 31;
  const int lo = lane & 15, hi = lane >> 4;
  const int b = blockIdx.y;
  const int p0 = blockIdx.x * (NW * 16) + wid * 16;

  const float* adj_row = adj + ((size_t)b * PP + p0 + lo) * CC;  // A: m=p, k=c contig
  const float* nc_b = new_channel + (size_t)b * CC * HH;

  v4f r0[4], r1[4];
  load_chunk(nc_b, tid, r0, r1);

  v8f acc0 = {}, acc1 = {};
  const int NCH = CC / CHUNK;  // 16
  for (int ci = 0; ci < NCH; ++ci) {
    float* buf = lds_buf[ci & 1];
    store_chunk(buf, tid, r0, r1);
    __syncthreads();
    if (ci + 1 < NCH)
      load_chunk(nc_b + (size_t)(ci + 1) * CHUNK * HH, tid, r0, r1);

    const int ch = ci * CHUNK;
#pragma unroll 8
    for (int k4 = 0; k4 < CHUNK / 4; ++k4) {
      // A[m][k] = adj[p0+m][c+k]: one contiguous 8B load per lane (global_load_b64)
      v2f a = *(const v2f*)(adj_row + ch + 4 * k4 + 2 * hi);
      const float* bp = buf + (2 * k4 + hi) * LSTRIDE + 2 * lo;
      v2f b0 = *(const v2f*)(bp);
      v2f b1 = *(const v2f*)(bp + 32);
      acc0 = wmma_f32(a, b0, acc0);
      acc1 = wmma_f32(a, b1, acc1);
    }
  }

  float* msg = (float*)lds_msg[wid];
#pragma unroll
  for (int r = 0; r < 8; ++r) {
    const int m = r + 8 * hi;
    msg[m * HH + lo] = acc0[r];
    msg[m * HH + 16 + lo] = acc1[r];
  }
  gru_tile(acc0, acc1, path_states + ((size_t)b * PP + p0) * HH, msg,
           w_ih_p, w_hh_p, b_ih_p, b_hh_p,
           new_path + ((size_t)b * PP + p0) * HH, lane);
}

extern "C" void kernel_launch(void* const* d_in, const int* in_sizes, int n_in,
                              void* d_out, int out_size, void* d_ws, size_t ws_size,
                              hipStream_t stream) {
  const float* path_states    = (const float*)d_in[0];
  const float* channel_states = (const float*)d_in[1];
  const float* adj            = (const float*)d_in[2];
  const float* w_ih_c = (const float*)d_in[3];
  const float* w_hh_c = (const float*)d_in[4];
  const float* b_ih_c = (const float*)d_in[5];
  const float* b_hh_c = (const float*)d_in[6];
  const float* w_ih_p = (const float*)d_in[7];
  const float* w_hh_p = (const float*)d_in[8];
  const float* b_ih_p = (const float*)d_in[9];
  const float* b_hh_p = (const float*)d_in[10];

  float* new_path    = (float*)d_out;                          // [B,P,H]
  float* new_channel = (float*)d_out + (size_t)BB * PP * HH;   // [B,C,H]

  dim3 grid1(CC / (NW * 16), BB);  // 32 x 16 workgroups
  dim3 grid2(PP / (NW * 16), BB);
  k_channel<<<grid1, NW * 32, 0, stream>>>(path_states, channel_states, adj,
                                           w_ih_c, w_hh_c, b_ih_c, b_hh_c,
                                           new_channel);
  k_path<<<grid2, NW * 32, 0, stream>>>(path_states, adj, new_channel,
                                        w_ih_p, w_hh_p, b_ih_p, b_hh_p,
                                        new_path);
}